// MyMaxPool2D_56444460204140
// MI455X (gfx1250) — compile-verified
//
#include <hip/hip_runtime.h>
#include <math.h>

typedef __attribute__((ext_vector_type(2))) float v2f;
typedef __attribute__((ext_vector_type(8))) float v8f;

#define NB   64     // batch
#define NC   512    // channels
#define HH   28
#define WW   28
#define WO   14
#define NPOS 196    // 14*14
#define PPAD 208    // 13*16 padded positions
#define NTILE 13
#define CG   64     // channels per group
#define NG   8      // channel groups
#define TOTAL (NB * NC * NPOS)

// ---------------------------------------------------------------------------
// Kernel A: fused 2x2/2 max-pool + per-(batch,group,pos) softmax statistics.
// Phase 1 (scalar): pool 64 channels x 196 positions, stage in LDS, track max.
// Phase 2 (WMMA):   S = sum_c exp(v-M), T = sum_c exp(v-M)*v via
//                   v_wmma_f32_16x16x4_f32 against an all-ones B matrix.
// ---------------------------------------------------------------------------
__global__ __launch_bounds__(256)
void pool_stats_kernel(const float* __restrict__ x,
                       float* __restrict__ pooled,           // [NB][NC][NPOS]
                       float* __restrict__ partial)          // [NB][NG][NPOS][3] = M,S,T
{
    __shared__ float tile[CG][PPAD];   // pooled values, [channel][position]
    __shared__ float mrow[PPAD];       // per-position max over this group

    const int g = blockIdx.x;          // channel group 0..7
    const int b = blockIdx.y;          // batch 0..63
    const int t = threadIdx.x;

    // ---- Phase 1: pooling ----
    if (t < NPOS) {
        const int ho = t / WO, wo = t % WO;
        const float* xb = x + (((size_t)b * NC + (size_t)g * CG) * HH + 2 * ho) * WW + 2 * wo;
        float* pb = pooled + ((size_t)b * NC + (size_t)g * CG) * NPOS + t;
        float m = -INFINITY;
        #pragma unroll 4
        for (int c = 0; c < CG; ++c) {
            const float* xc = xb + (size_t)c * (HH * WW);
            v2f r0 = *(const v2f*)(xc);        // 8B-aligned: element index is even
            v2f r1 = *(const v2f*)(xc + WW);
            float v = fmaxf(fmaxf(r0.x, r0.y), fmaxf(r1.x, r1.y));
            pb[(size_t)c * NPOS] = v;
            tile[c][t] = v;
            m = fmaxf(m, v);
        }
        mrow[t] = m;
    } else if (t < PPAD) {
        // Pad columns: exp(-1e30 - 0) == 0, so padded rows contribute nothing
        // (and WMMA rows are independent anyway).
        mrow[t] = 0.0f;
        for (int c = 0; c < CG; ++c) tile[c][t] = -1e30f;
    }
    __syncthreads();

    // ---- Phase 2: WMMA channel reduction ----
    const int lane = t & 31;
    const int wave = t >> 5;           // 8 waves per block
    const int half = lane >> 4;        // 0: K=0,1   1: K=2,3
    const int m16  = lane & 15;        // row within 16x16 tile

    v2f onesB; onesB.x = 1.0f; onesB.y = 1.0f;   // all-ones B: layout-independent

    for (int tl = wave; tl < NTILE; tl += 8) {   // uniform per wave -> EXEC all 1s
        const int p0 = tl * 16;
        const int p  = p0 + m16;
        const float M = mrow[p];

        v8f accS = {};   // per-row sum of exp(v - M)
        v8f accT = {};   // per-row sum of exp(v - M) * v

        #pragma unroll
        for (int kk = 0; kk < CG / 4; ++kk) {
            const int c0 = kk * 4 + half * 2;
            const float v0 = tile[c0 + 0][p];
            const float v1 = tile[c0 + 1][p];
            const float e0 = expf(v0 - M);
            const float e1 = expf(v1 - M);
            v2f aE; aE.x = e0;      aE.y = e1;        // A[m][K], A[m][K+1]
            v2f aT; aT.x = e0 * v0; aT.y = e1 * v1;
            accS = __builtin_amdgcn_wmma_f32_16x16x4_f32(
                       false, aE, false, onesB, (short)0, accS, false, false);
            accT = __builtin_amdgcn_wmma_f32_16x16x4_f32(
                       false, aT, false, onesB, (short)0, accT, false, false);
        }

        // Extract: VGPR r holds row r (lanes 0-15) / row r+8 (lanes 16-31);
        // all 16 columns of a row are identical, use column == r as writer.
        #pragma unroll
        for (int r = 0; r < 8; ++r) {
            const int prow = p0 + r + half * 8;
            if (m16 == r && prow < NPOS) {
                float* dst = partial + (((size_t)b * NG + g) * NPOS + prow) * 3;
                dst[0] = mrow[prow];
                dst[1] = accS[r];
                dst[2] = accT[r];
            }
        }
    }
}

// ---------------------------------------------------------------------------
// Kernel B: combine the 8 group partials per position (online-softmax merge),
// compute entropy, per-batch spatial max, emit weight = 1 - ent/max_ent.
// ---------------------------------------------------------------------------
__global__ __launch_bounds__(256)
void entropy_weight_kernel(const float* __restrict__ partial,  // [NB][NG][NPOS][3]
                           float* __restrict__ weight)         // [NB][NPOS]
{
    __shared__ float sred[256];
    const int b = blockIdx.x;
    const int t = threadIdx.x;

    float entv = 0.0f;
    float entmax_in = -INFINITY;

    if (t < NPOS) {
        float Mg[NG], Sg[NG], Tg[NG];
        float M = -INFINITY;
        #pragma unroll
        for (int g = 0; g < NG; ++g) {
            const float* src = partial + (((size_t)b * NG + g) * NPOS + t) * 3;
            Mg[g] = src[0]; Sg[g] = src[1]; Tg[g] = src[2];
            M = fmaxf(M, Mg[g]);
        }
        float S = 0.0f, T = 0.0f;
        #pragma unroll
        for (int g = 0; g < NG; ++g) {
            const float sc = expf(Mg[g] - M);
            S += Sg[g] * sc;
            T += Tg[g] * sc;
        }
        // ent = -sum p*logp = M + log(S) - T/S
        entv = M + logf(S) - T / S;
        entmax_in = entv;
    }
    sred[t] = entmax_in;
    __syncthreads();
    #pragma unroll
    for (int s = 128; s > 0; s >>= 1) {
        if (t < s) sred[t] = fmaxf(sred[t], sred[t + s]);
        __syncthreads();
    }
    const float maxent = sred[0];

    if (t < NPOS) {
        weight[(size_t)b * NPOS + t] = 1.0f - entv / maxent;
    }
}

// ---------------------------------------------------------------------------
// Kernel C: in-place broadcast scale of the pooled output.
// ---------------------------------------------------------------------------
__global__ __launch_bounds__(256)
void scale_kernel(float* __restrict__ out, const float* __restrict__ weight)
{
    const unsigned i = blockIdx.x * 256u + threadIdx.x;
    if (i < (unsigned)TOTAL) {
        const unsigned pos = i % NPOS;
        const unsigned b   = i / (NC * NPOS);
        out[i] *= weight[b * NPOS + pos];
    }
}

extern "C" void kernel_launch(void* const* d_in, const int* in_sizes, int n_in,
                              void* d_out, int out_size, void* d_ws, size_t ws_size,
                              hipStream_t stream)
{
    (void)in_sizes; (void)n_in; (void)out_size; (void)ws_size;
    const float* x = (const float*)d_in[0];
    float* out = (float*)d_out;

    float* partial = (float*)d_ws;                                   // NB*NG*NPOS*3 floats
    float* weight  = partial + (size_t)NB * NG * NPOS * 3;           // NB*NPOS floats

    pool_stats_kernel<<<dim3(NG, NB), 256, 0, stream>>>(x, out, partial);
    entropy_weight_kernel<<<NB, 256, 0, stream>>>(partial, weight);
    scale_kernel<<<(TOTAL + 255) / 256, 256, 0, stream>>>(out, weight);
}